// GraphEncoder_55301998903767
// MI455X (gfx1250) — compile-verified
//
#include <hip/hip_runtime.h>
#include <hip/hip_bf16.h>
#include <math.h>

// ---------------------------------------------------------------------------
// GraphEncoder for MI455X (gfx1250, wave32).
// All dense GEMMs use V_WMMA_F32_16X16X4_F32 (full fp32 matrix path) since the
// reference is fp32 and stacks 4 LayerNorms (error amplifiers).
// Per ISA 7.12.2 (wave32):
//   A (16x4 f32, 2 VGPR):  A[m][k] -> vgpr (k&1), lane = m + 16*(k>>1)
//   B ( 4x16 f32, 2 VGPR): B[k][n] -> vgpr (k&1), lane = n + 16*(k>>1)
//   C/D (16x16 f32, 8 VGPR): D[m][n] -> vgpr (m&7), lane = n + 16*(m>>3)
// A-tiles are staged through LDS (coalesced global load once per block,
// conflict-free ds reads via odd row stride) instead of 512B-strided
// per-fragment global loads repeated by every wave.
// ---------------------------------------------------------------------------

typedef __attribute__((ext_vector_type(2))) float v2f;
typedef __attribute__((ext_vector_type(8))) float v8f;

#define Bn   1024
#define Nn   512
#define Fn   17
#define Dn   128
#define HCn  96
#define On   256

__device__ __forceinline__ v8f wmma4(v2f a, v2f b, v8f c) {
  // D = A(16x4,f32) * B(4x16,f32) + C ; emits v_wmma_f32_16x16x4_f32
  return __builtin_amdgcn_wmma_f32_16x16x4_f32(
      /*neg_a=*/false, a, /*neg_b=*/false, b,
      /*c_mod=*/(short)0, c, /*reuse_a=*/false, /*reuse_b=*/false);
}

__device__ __forceinline__ float gelu_exact(float x) {
  return 0.5f * x * (1.0f + erff(x * 0.70710678118654752f));
}

// ---------------------------------------------------------------------------
// k1: y1[b,n,0:128] = LN( GELU( (x * x[...,16]) @ W1 + b1 ) )
// grid = B * N/16 blocks, 128 threads (4 waves). wave w -> cols [32w,32w+32).
// Raw x rows staged in LDS, K zero-padded 17 -> 20 there (no divergence in
// the WMMA loop); gating (row * row[16]) applied on the A-fragment read.
// ---------------------------------------------------------------------------
__global__ __launch_bounds__(128) void k1_in_gemm(
    const float* __restrict__ x, const float* __restrict__ W1,
    const float* __restrict__ b1, const float* __restrict__ lng,
    const float* __restrict__ lnb, float* __restrict__ y1) {
  __shared__ float xt[16][20];       // stride 20 -> 16 distinct banks per col
  __shared__ float tile[16][Dn];
  __shared__ float mu[16], rs[16];
  int bid = blockIdx.x;
  int b = bid >> 5, n0 = (bid & 31) << 4;
  int tid = threadIdx.x;
  int wave = tid >> 5, lane = tid & 31, ll = lane & 15, hf = lane >> 4;

  // cooperative coalesced stage of the 16x17 tile (+ zero pad to 20)
  for (int idx = tid; idx < 16 * 20; idx += 128) {
    int r = idx / 20, c = idx % 20;
    xt[r][c] = (c < Fn) ? x[(size_t)(b * Nn + n0 + r) * Fn + c] : 0.f;
  }
  __syncthreads();

  float x16 = xt[ll][16];            // visibility channel of this lane's row
  int c0 = wave * 32;
  v8f acc0 = {}; v8f acc1 = {};
#pragma unroll
  for (int kk = 0; kk < 5; ++kk) {
    v2f a, w0, w1;
#pragma unroll
    for (int v = 0; v < 2; ++v) {
      int k = kk * 4 + hf * 2 + v;
      a[v]  = xt[ll][k] * x16;                         // gated A fragment
      w0[v] = (k < Fn) ? W1[k * Dn + c0 + ll]      : 0.f;
      w1[v] = (k < Fn) ? W1[k * Dn + c0 + 16 + ll] : 0.f;
    }
    acc0 = wmma4(a, w0, acc0);
    acc1 = wmma4(a, w1, acc1);
  }
#pragma unroll
  for (int r = 0; r < 8; ++r) {
    int m = r + hf * 8;
    tile[m][c0 + ll]      = gelu_exact(acc0[r] + b1[c0 + ll]);
    tile[m][c0 + 16 + ll] = gelu_exact(acc1[r] + b1[c0 + 16 + ll]);
  }
  __syncthreads();
  if (tid < 16) {
    float s = 0.f;
    for (int j = 0; j < Dn; ++j) s += tile[tid][j];
    float m_ = s * (1.f / Dn);
    float v_ = 0.f;
    for (int j = 0; j < Dn; ++j) { float d = tile[tid][j] - m_; v_ += d * d; }
    mu[tid] = m_;
    rs[tid] = rsqrtf(v_ * (1.f / Dn) + 1e-6f);
  }
  __syncthreads();
  for (int i = 0; i < 16; ++i) {
    y1[(size_t)(b * Nn + n0 + i) * Dn + tid] =
        (tile[i][tid] - mu[i]) * rs[i] * lng[tid] + lnb[tid];
  }
}

// ---------------------------------------------------------------------------
// k_xl0: xl0[b,hc] = yin[b,0,:DIN] . Wl[:,hc] + bl[hc]   (only node 0 used)
// grid = B blocks, 96 threads. DIN compile-time -> unrolled by 8.
// ---------------------------------------------------------------------------
template <int DIN>
__global__ __launch_bounds__(96) void k_xl0(
    const float* __restrict__ yin, const float* __restrict__ Wl,
    const float* __restrict__ bl, float* __restrict__ xl0) {
  int b = blockIdx.x, hc = threadIdx.x;
  const float* r = yin + (size_t)b * Nn * DIN;
  float s = bl[hc];
#pragma unroll 8
  for (int k = 0; k < DIN; ++k) s += r[k] * Wl[k * HCn + hc];
  xl0[b * HCn + hc] = s;
}

// ---------------------------------------------------------------------------
// k_xr: x_r[b,n,:] = yin[b,n,:DIN] @ Wr + br (WMMA), plus attention logits
//   logits[b,n,h] = sum_c leakyrelu(xl0[b,h,c] + x_r[b,n,h,c]) * att[h,c]
// grid = B * N/16 blocks, 192 threads (6 waves, one 16-col tile each).
// A tile staged once in LDS (stride DIN+1 -> conflict-free fragment reads).
// ---------------------------------------------------------------------------
template <int DIN>
__global__ __launch_bounds__(192) void k_xr(
    const float* __restrict__ yin, const float* __restrict__ Wr,
    const float* __restrict__ br, const float* __restrict__ att,
    const float* __restrict__ xl0, float* __restrict__ xr,
    float* __restrict__ logits) {
  __shared__ float at[16][DIN + 1];  // odd stride -> no bank conflicts
  __shared__ float tile[16][HCn];
  int bid = blockIdx.x;
  int b = bid >> 5, n0 = (bid & 31) << 4;
  int tid = threadIdx.x;
  int wave = tid >> 5, lane = tid & 31, ll = lane & 15, hf = lane >> 4;
  int c0 = wave * 16;

  // cooperative coalesced stage of the 16 x DIN activation tile
  const float* src = yin + (size_t)(b * Nn + n0) * DIN;
  for (int idx = tid; idx < 16 * DIN; idx += 192) {
    int r = idx / DIN, c = idx % DIN;
    at[r][c] = src[(size_t)r * DIN + c];
  }
  __syncthreads();

  v8f acc = {};
#pragma unroll
  for (int k0 = 0; k0 < DIN; k0 += 4) {
    v2f a, w;
#pragma unroll
    for (int v = 0; v < 2; ++v) {
      int k = k0 + hf * 2 + v;
      a[v] = at[ll][k];
      w[v] = Wr[k * HCn + c0 + ll];
    }
    acc = wmma4(a, w, acc);
  }
#pragma unroll
  for (int r = 0; r < 8; ++r) {
    int m = r + hf * 8;
    float val = acc[r] + br[c0 + ll];
    tile[m][c0 + ll] = val;
    xr[(size_t)(b * Nn + n0 + m) * HCn + c0 + ll] = val;
  }
  __syncthreads();
  if (tid < 48) {                    // 16 rows x 3 heads
    int rw = tid / 3, h = tid % 3;
    const float* xl = xl0 + b * HCn + h * 32;
    const float* atp = att + h * 32;
    float lg = 0.f;
#pragma unroll
    for (int c = 0; c < 32; ++c) {
      float e = xl[c] + tile[rw][h * 32 + c];
      e = (e > 0.f) ? e : 0.2f * e;  // leaky_relu, slope 0.2
      lg += e * atp[c];
    }
    logits[(size_t)(b * Nn + n0 + rw) * 3 + h] = lg;
  }
}

// ---------------------------------------------------------------------------
// k_soft: per (b,h): alpha = softmax_n(logits); m_center[c] = sum_n alpha*xr;
// overwrite xr[b,0,h*32+c] with m_center (columns disjoint across heads).
// grid = B*3 blocks, 256 threads.
// ---------------------------------------------------------------------------
__global__ __launch_bounds__(256) void k_soft(
    const float* __restrict__ logits, float* __restrict__ xr) {
  __shared__ float red[256];
  __shared__ float wl[Nn];
  __shared__ float part[256];
  int b = blockIdx.x / 3, h = blockIdx.x % 3;
  int t = threadIdx.x;
  float l0 = logits[((size_t)b * Nn + t) * 3 + h];
  float l1 = logits[((size_t)b * Nn + t + 256) * 3 + h];
  red[t] = fmaxf(l0, l1);
  __syncthreads();
  for (int s = 128; s > 0; s >>= 1) {
    if (t < s) red[t] = fmaxf(red[t], red[t + s]);
    __syncthreads();
  }
  float gmax = red[0];
  __syncthreads();
  float e0 = expf(l0 - gmax), e1 = expf(l1 - gmax);
  wl[t] = e0; wl[t + 256] = e1;
  red[t] = e0 + e1;
  __syncthreads();
  for (int s = 128; s > 0; s >>= 1) {
    if (t < s) red[t] += red[t + s];
    __syncthreads();
  }
  float Z = red[0];
  // weighted mean of x_r over nodes: 8 chunks of 64 nodes x 32 channels
  int c = t & 31, chunk = t >> 5;
  const float* base = xr + (size_t)b * Nn * HCn + h * 32 + c;
  float s_ = 0.f;
  for (int n = chunk * 64; n < chunk * 64 + 64; ++n)
    s_ += wl[n] * base[(size_t)n * HCn];
  part[t] = s_;
  __syncthreads();
  if (t < 32) {
    float a = 0.f;
#pragma unroll
    for (int j = 0; j < 8; ++j) a += part[j * 32 + t];
    xr[(size_t)b * Nn * HCn + h * 32 + t] = a / Z;   // replace node 0
  }
}

// ---------------------------------------------------------------------------
// k_geluln: out[row,:] = LN(GELU(in[row,:])) over HC=96. grid=B*N, 96 thr.
// ---------------------------------------------------------------------------
__global__ __launch_bounds__(96) void k_geluln(
    const float* __restrict__ in, const float* __restrict__ g,
    const float* __restrict__ bb, float* __restrict__ out) {
  __shared__ float red[96];
  size_t row = blockIdx.x;
  int t = threadIdx.x;
  float v = gelu_exact(in[row * HCn + t]);
  red[t] = v;
  __syncthreads();
  if (t < 32) red[t] += red[t + 32] + red[t + 64];
  __syncthreads();
  for (int s = 16; s > 0; s >>= 1) { if (t < s) red[t] += red[t + s]; __syncthreads(); }
  float mu_ = red[0] * (1.f / HCn);
  __syncthreads();
  float d = v - mu_;
  red[t] = d * d;
  __syncthreads();
  if (t < 32) red[t] += red[t + 32] + red[t + 64];
  __syncthreads();
  for (int s = 16; s > 0; s >>= 1) { if (t < s) red[t] += red[t + s]; __syncthreads(); }
  float rs_ = rsqrtf(red[0] * (1.f / HCn) + 1e-6f);
  out[row * HCn + t] = d * rs_ * g[t] + bb[t];
}

// ---------------------------------------------------------------------------
// k_pool: per (b, feature): temperature-softmax weighted sum over 512 nodes.
// grid = B blocks, 96 threads.
// ---------------------------------------------------------------------------
__global__ __launch_bounds__(96) void k_pool(
    const float* __restrict__ y3, const float* __restrict__ tptr,
    float* __restrict__ pooled) {
  int b = blockIdx.x, t = threadIdx.x;
  float invt = 1.f / (tptr[0] + 1e-8f);
  const float* base = y3 + (size_t)b * Nn * HCn + t;
  float mx = -3.4e38f;
  for (int n = 0; n < Nn; ++n) mx = fmaxf(mx, base[(size_t)n * HCn] * invt);
  float Z = 0.f, S = 0.f;
  for (int n = 0; n < Nn; ++n) {
    float v = base[(size_t)n * HCn];
    float e = expf(v * invt - mx);
    Z += e; S += e * v;
  }
  pooled[b * HCn + t] = S / Z;
}

// ---------------------------------------------------------------------------
// k_final: out = LN(GELU(pooled @ Wo + bo)) ; [1024,96]@[96,256].
// grid = B/16 = 64 blocks, 512 threads (16 waves, one 16-col tile each).
// Shared A tile staged once in LDS (was re-loaded by all 16 waves).
// ---------------------------------------------------------------------------
__global__ __launch_bounds__(512) void k_final(
    const float* __restrict__ pooled, const float* __restrict__ Wo,
    const float* __restrict__ bo, const float* __restrict__ g,
    const float* __restrict__ bb, float* __restrict__ out) {
  __shared__ float at[16][HCn + 1];  // odd stride -> conflict-free
  __shared__ float tile[16][On];
  __shared__ float mu[16], rs[16];
  int b0 = blockIdx.x * 16;
  int tid = threadIdx.x;
  int wave = tid >> 5, lane = tid & 31, ll = lane & 15, hf = lane >> 4;
  int c0 = wave * 16;

  for (int idx = tid; idx < 16 * HCn; idx += 512) {
    int r = idx / HCn, c = idx % HCn;
    at[r][c] = pooled[(size_t)(b0 + r) * HCn + c];
  }
  __syncthreads();

  v8f acc = {};
#pragma unroll
  for (int k0 = 0; k0 < HCn; k0 += 4) {
    v2f a, w;
#pragma unroll
    for (int v = 0; v < 2; ++v) {
      int k = k0 + hf * 2 + v;
      a[v] = at[ll][k];
      w[v] = Wo[k * On + c0 + ll];
    }
    acc = wmma4(a, w, acc);
  }
#pragma unroll
  for (int r = 0; r < 8; ++r) {
    int m = r + hf * 8;
    tile[m][c0 + ll] = gelu_exact(acc[r] + bo[c0 + ll]);
  }
  __syncthreads();
  if (tid < 16) {
    float s = 0.f;
    for (int j = 0; j < On; ++j) s += tile[tid][j];
    float m_ = s * (1.f / On);
    float v_ = 0.f;
    for (int j = 0; j < On; ++j) { float d = tile[tid][j] - m_; v_ += d * d; }
    mu[tid] = m_;
    rs[tid] = rsqrtf(v_ * (1.f / On) + 1e-6f);
  }
  __syncthreads();
#pragma unroll
  for (int i = 0; i < 8; ++i) {
    int idx = tid + 512 * i;
    int rw = idx >> 8, col = idx & 255;
    out[(size_t)(b0 + rw) * On + col] =
        (tile[rw][col] - mu[rw]) * rs[rw] * g[col] + bb[col];
  }
}

// ---------------------------------------------------------------------------
extern "C" void kernel_launch(void* const* d_in, const int* in_sizes, int n_in,
                              void* d_out, int out_size, void* d_ws,
                              size_t ws_size, hipStream_t stream) {
  const float* x     = (const float*)d_in[0];
  const float* W1    = (const float*)d_in[1];
  const float* b1    = (const float*)d_in[2];
  const float* ln1g  = (const float*)d_in[3];
  const float* ln1b  = (const float*)d_in[4];
  const float* g1Wl  = (const float*)d_in[5];
  const float* g1bl  = (const float*)d_in[6];
  const float* g1Wr  = (const float*)d_in[7];
  const float* g1br  = (const float*)d_in[8];
  const float* g1att = (const float*)d_in[9];
  const float* ln2g  = (const float*)d_in[10];
  const float* ln2b  = (const float*)d_in[11];
  const float* g2Wl  = (const float*)d_in[12];
  const float* g2bl  = (const float*)d_in[13];
  const float* g2Wr  = (const float*)d_in[14];
  const float* g2br  = (const float*)d_in[15];
  const float* g2att = (const float*)d_in[16];
  const float* ln3g  = (const float*)d_in[17];
  const float* ln3b  = (const float*)d_in[18];
  const float* tsc   = (const float*)d_in[19];
  const float* Wo    = (const float*)d_in[20];
  const float* bo    = (const float*)d_in[21];
  const float* ln4g  = (const float*)d_in[22];
  const float* ln4b  = (const float*)d_in[23];
  float* out = (float*)d_out;

  // workspace carve-up (floats)
  float* buf1   = (float*)d_ws;                       // [B,N,128] then [B,N,96]
  float* buf2   = buf1 + (size_t)Bn * Nn * Dn;        // [B,N,96] x_r / gat out
  float* lg     = buf2 + (size_t)Bn * Nn * HCn;       // [B,N,3] logits
  float* xl0    = lg   + (size_t)Bn * Nn * 3;         // [B,96]
  float* pooled = xl0  + (size_t)Bn * HCn;            // [B,96]

  dim3 gridTile(Bn * (Nn / 16));

  // stage 1: gated input GEMM + GELU + LN  -> buf1 [B,N,128]
  k1_in_gemm<<<gridTile, 128, 0, stream>>>(x, W1, b1, ln1g, ln1b, buf1);

  // GAT layer 1 (Din = 128)
  k_xl0<Dn><<<Bn, HCn, 0, stream>>>(buf1, g1Wl, g1bl, xl0);
  k_xr<Dn><<<gridTile, 192, 0, stream>>>(buf1, g1Wr, g1br, g1att, xl0, buf2, lg);
  k_soft<<<Bn * 3, 256, 0, stream>>>(lg, buf2);
  k_geluln<<<Bn * Nn, HCn, 0, stream>>>(buf2, ln2g, ln2b, buf1);  // buf1 now [B,N,96]

  // GAT layer 2 (Din = 96)
  k_xl0<HCn><<<Bn, HCn, 0, stream>>>(buf1, g2Wl, g2bl, xl0);
  k_xr<HCn><<<gridTile, 192, 0, stream>>>(buf1, g2Wr, g2br, g2att, xl0, buf2, lg);
  k_soft<<<Bn * 3, 256, 0, stream>>>(lg, buf2);
  k_geluln<<<Bn * Nn, HCn, 0, stream>>>(buf2, ln3g, ln3b, buf1);

  // pooling + output head
  k_pool<<<Bn, HCn, 0, stream>>>(buf1, tsc, pooled);
  k_final<<<Bn / 16, 512, 0, stream>>>(pooled, Wo, bo, ln4g, ln4b, out);
}